// KAN_89481348645393
// MI455X (gfx1250) — compile-verified
//
#include <hip/hip_runtime.h>
#include <stddef.h>

typedef float v2f __attribute__((ext_vector_type(2)));
typedef float v4f __attribute__((ext_vector_type(4)));
typedef float v8f __attribute__((ext_vector_type(8)));
typedef int   v4i __attribute__((vector_size(16)));   // matches builtin's V4i

#ifndef __has_builtin
#define __has_builtin(x) 0
#endif

#if __has_builtin(__builtin_amdgcn_global_load_async_to_lds_b128)
#define ASYNC_LDS 1
#else
#define ASYNC_LDS 0
#endif

#define NROWS 65536
#define DDIM  1024
#define NBR   256
#define NOUTC 512
#define MBLK  32
#define KC    64
#define NKC   (DDIM / KC)
#define XS_STRIDE 68   // 272 B row stride: 16B-aligned, conflict-free A-frag reads
#define HS_STRIDE 260  // pad 256->260: conflict-free column-fragment access

__device__ __forceinline__ void async_wait_all() {
#if ASYNC_LDS
#if __has_builtin(__builtin_amdgcn_s_wait_asynccnt)
  __builtin_amdgcn_s_wait_asynccnt(0);
#else
  asm volatile("s_wait_asynccnt 0" ::: "memory");
#endif
#endif
}

__device__ __forceinline__ void stage16(const float* __restrict__ g,
                                        float* __restrict__ l) {
#if ASYNC_LDS
  __builtin_amdgcn_global_load_async_to_lds_b128(
      (__attribute__((address_space(1))) v4i*)(g),
      (__attribute__((address_space(3))) v4i*)(l), 0, 0);
#else
  *(v4f*)l = *(const v4f*)g;
#endif
}

// stage one 32x64 f32 x-chunk: 512 float4s over 256 threads -> 2 async b128 each
__device__ __forceinline__ void stage_chunk(const float* __restrict__ gbase,
                                            float (*__restrict__ xs)[XS_STRIDE],
                                            int sr, int sc) {
  stage16(gbase,      &xs[sr][sc]);
  stage16(gbase + 32, &xs[sr][sc + 32]);
}

__global__ __launch_bounds__(256)
void moe_fused_kernel(const float* __restrict__ xg,
                      const float* __restrict__ Wg,
                      const float* __restrict__ bg,
                      const float* __restrict__ aggg,
                      float* __restrict__ outg) {
  __shared__ float xs[2][MBLK][XS_STRIDE];  // 17408 B, double-buffered x chunk
  __shared__ float hs[MBLK][HS_STRIDE];     // 33280 B, activated H block

  const int tid    = threadIdx.x;
  const int wave   = tid >> 5;
  const int lane   = tid & 31;
  const int lane16 = lane & 15;
  const int khalf  = (lane >> 4) << 1;  // 0 or 2: K offset of this half-wave (A/B frag)
  const int rhalf  = (lane >> 4) << 3;  // 0 or 8: M offset of this half-wave (C/D frag)
  const int rt     = wave >> 2;         // row tile 0..1
  const int cg     = wave & 3;          // column group 0..3 (== activation block, stage 1)
  const int rowBase = blockIdx.x * MBLK;

  // cooperative x-chunk staging map: 256 threads cover 32 rows x 8 float4s (x2 halves)
  const int sr = tid >> 3;
  const int sc = (tid & 7) << 2;
  const float* gsrc_base = xg + (size_t)(rowBase + sr) * DDIM + sc;

  // ---------------- Stage 1: H = x @ W  (K = 1024) ----------------
  stage_chunk(gsrc_base, xs[0], sr, sc);
  async_wait_all();
  __syncthreads();

  v8f acc[4];
#pragma unroll
  for (int t = 0; t < 4; ++t) acc[t] = (v8f){0, 0, 0, 0, 0, 0, 0, 0};

  for (int kc = 0; kc < NKC; ++kc) {
    const int cur = kc & 1;
    if (kc + 1 < NKC)  // prefetch next chunk while computing on current
      stage_chunk(gsrc_base + (size_t)(kc + 1) * KC, xs[cur ^ 1], sr, sc);

    const int kbase = kc * KC;
#pragma unroll 4
    for (int kk = 0; kk < KC; kk += 4) {
      // A fragment (16x4 f32): lanes 0-15 hold K=0,1; lanes 16-31 hold K=2,3
      v2f a = *(const v2f*)&xs[cur][(rt << 4) + lane16][kk + khalf];
      const float* wcol = Wg + (size_t)(kbase + kk + khalf) * NBR + lane16;
#pragma unroll
      for (int t = 0; t < 4; ++t) {
        const int nof = (((cg << 2) + t) << 4);
        v2f bf;
        bf.x = wcol[nof];
        bf.y = wcol[NBR + nof];
        acc[t] = __builtin_amdgcn_wmma_f32_16x16x4_f32(
            false, a, false, bf, (short)0, acc[t], false, false);
      }
    }
    async_wait_all();
    __syncthreads();
  }

  // ---------------- bias + per-block activation -> LDS ----------------
  const float inv_sqrt2 = 0.70710678118654752f;
#pragma unroll
  for (int t = 0; t < 4; ++t) {
    const int ct  = (cg << 2) + t;
    const int col = (ct << 4) + lane16;
    const float bias = bg[col];
#pragma unroll
    for (int v = 0; v < 8; ++v) {
      float h = acc[t][v] + bias;
      float a;
      if (cg == 0)      a = fmaxf(h, 0.0f);                          // relu
      else if (cg == 1) a = tanhf(h);                                // tanh
      else if (cg == 2) a = 1.0f / (1.0f + expf(-h));                // sigmoid
      else              a = 0.5f * h * (1.0f + erff(h * inv_sqrt2)); // exact gelu
      hs[(rt << 4) + v + rhalf][col] = a;
    }
  }
  __syncthreads();

  // ---------------- Stage 2: out = act(H) @ agg  (K = 256) ----------------
  v8f oacc[8];
#pragma unroll
  for (int t = 0; t < 8; ++t) oacc[t] = (v8f){0, 0, 0, 0, 0, 0, 0, 0};

#pragma unroll 2
  for (int k = 0; k < NBR; k += 4) {
    v2f a = *(const v2f*)&hs[(rt << 4) + lane16][k + khalf];
    const float* acol = aggg + (size_t)(k + khalf) * NOUTC + lane16;
#pragma unroll
    for (int t = 0; t < 8; ++t) {
      const int nof = (((cg << 3) + t) << 4);
      v2f bf;
      bf.x = acol[nof];
      bf.y = acol[NOUTC + nof];
      oacc[t] = __builtin_amdgcn_wmma_f32_16x16x4_f32(
          false, a, false, bf, (short)0, oacc[t], false, false);
    }
  }

  // ---------------- store D tiles ----------------
  float* orow = outg + (size_t)(rowBase + (rt << 4) + rhalf) * NOUTC + lane16;
#pragma unroll
  for (int v = 0; v < 8; ++v) {
#pragma unroll
    for (int t = 0; t < 8; ++t) {
      const int nof = (((cg << 3) + t) << 4);
      orow[(size_t)v * NOUTC + nof] = oacc[t][v];
    }
  }
}

extern "C" void kernel_launch(void* const* d_in, const int* in_sizes, int n_in,
                              void* d_out, int out_size, void* d_ws, size_t ws_size,
                              hipStream_t stream) {
  (void)in_sizes; (void)n_in; (void)d_ws; (void)ws_size; (void)out_size;
  const float* x   = (const float*)d_in[0];  // [65536,1024]
  const float* W   = (const float*)d_in[1];  // [1024,256]
  const float* b   = (const float*)d_in[2];  // [256]
  const float* agg = (const float*)d_in[3];  // [256,512]
  float* out = (float*)d_out;                // [65536,512]

  dim3 grid(NROWS / MBLK);  // 2048 workgroups
  dim3 block(256);          // 8 wave32s
  moe_fused_kernel<<<grid, block, 0, stream>>>(x, W, b, agg, out);
}